// DensitySetAbstraction_3281355014264
// MI455X (gfx1250) — compile-verified
//
#include <hip/hip_runtime.h>
#include <hip/hip_bf16.h>
#include <math.h>

// ---------------------------------------------------------------------------
// PointNet++ set abstraction for MI455X (gfx1250, wave32)
//   k1: farthest point sampling (1 block per batch, dists in registers)
//   k2: ball query + group + center-subtract (1 wave per centroid, ballot scan)
//   k3: MLP (3->64->64->128, folded BN+ReLU) via V_WMMA_F32_16X16X4_F32 + maxpool
// ---------------------------------------------------------------------------

constexpr int   BB   = 8;      // batches
constexpr int   NN   = 8192;   // input points
constexpr int   MM   = 2048;   // sampled points (NPOINT)
constexpr int   NS   = 32;     // samples per ball (NSAMPLE)
constexpr float R2   = 0.2f * 0.2f;
constexpr float EPSV = 1e-5f;

typedef __attribute__((ext_vector_type(2))) float v2f;
typedef __attribute__((ext_vector_type(8))) float v8f;

__device__ __forceinline__ v8f wmma_k4(v2f a, v2f b, v8f c) {
  // D = A(16x4,f32) * B(4x16,f32) + C(16x16,f32)
  return __builtin_amdgcn_wmma_f32_16x16x4_f32(
      /*neg_a=*/false, a, /*neg_b=*/false, b,
      /*c_mod=*/(short)0, c, /*reuse_a=*/false, /*reuse_b=*/false);
}

// ---------------------------------------------------------------------------
// Kernel 1: farthest point sampling. grid = B, block = 1024 (32 waves).
// Each thread owns 8 points (coords + running min-dist in registers).
// ---------------------------------------------------------------------------
__global__ __launch_bounds__(1024) void fps_kernel(
    const float* __restrict__ xyz, float* __restrict__ new_xyz) {
  const int b = blockIdx.x;
  const int t = threadIdx.x;
  const float* pts = xyz + (size_t)b * NN * 3;
  float* nx = new_xyz + (size_t)b * MM * 3;

  __shared__ float red_v[32];
  __shared__ int   red_j[32];
  __shared__ float wc[3];

  float px[8], py[8], pz[8], d[8];
#pragma unroll
  for (int k = 0; k < 8; ++k) {
    int j = t + k * 1024;
    px[k] = pts[j * 3 + 0];
    py[k] = pts[j * 3 + 1];
    pz[k] = pts[j * 3 + 2];
  }
  float cx = pts[0], cy = pts[1], cz = pts[2];
  if (t == 0) { nx[0] = cx; nx[1] = cy; nx[2] = cz; }
#pragma unroll
  for (int k = 0; k < 8; ++k) {
    float dx = px[k] - cx, dy = py[k] - cy, dz = pz[k] - cz;
    d[k] = dx * dx + dy * dy + dz * dz;
  }
  if (t == 0) d[0] = 0.0f;  // first centroid is index 0

  const int lane = t & 31, wv = t >> 5;
  for (int s = 1; s < MM; ++s) {
    // local argmax (within a thread, j increases with k -> strict > keeps first)
    float bv = d[0]; int bj = t;
#pragma unroll
    for (int k = 1; k < 8; ++k) {
      int j = t + k * 1024;
      if (d[k] > bv) { bv = d[k]; bj = j; }
    }
    // wave32 reduction (tie-break: lowest index, matching jnp.argmax)
#pragma unroll
    for (int off = 16; off > 0; off >>= 1) {
      float ov = __shfl_xor(bv, off);
      int   oj = __shfl_xor(bj, off);
      if (ov > bv || (ov == bv && oj < bj)) { bv = ov; bj = oj; }
    }
    if (lane == 0) { red_v[wv] = bv; red_j[wv] = bj; }
    __syncthreads();
    if (t < 32) {
      bv = red_v[t]; bj = red_j[t];
#pragma unroll
      for (int off = 16; off > 0; off >>= 1) {
        float ov = __shfl_xor(bv, off);
        int   oj = __shfl_xor(bj, off);
        if (ov > bv || (ov == bv && oj < bj)) { bv = ov; bj = oj; }
      }
      if (t == 0) {
        float qx = pts[bj * 3 + 0], qy = pts[bj * 3 + 1], qz = pts[bj * 3 + 2];
        wc[0] = qx; wc[1] = qy; wc[2] = qz;
        red_j[0] = bj;
        nx[s * 3 + 0] = qx; nx[s * 3 + 1] = qy; nx[s * 3 + 2] = qz;
      }
    }
    __syncthreads();
    float qx = wc[0], qy = wc[1], qz = wc[2];
    int win = red_j[0];
#pragma unroll
    for (int k = 0; k < 8; ++k) {
      int j = t + k * 1024;
      float dx = px[k] - qx, dy = py[k] - qy, dz = pz[k] - qz;
      float dd = dx * dx + dy * dy + dz * dz;
      d[k] = fminf(d[k], dd);
      if (j == win) d[k] = 0.0f;
    }
  }
}

// ---------------------------------------------------------------------------
// Kernel 2: ball query + grouping. One wave32 per centroid; ballot + prefix
// popc gives in-radius neighbor indices in ascending order (matches the
// reference's sort). Early-exit once 32 found. Writes centered coords to ws.
// ---------------------------------------------------------------------------
__global__ __launch_bounds__(256) void ballgroup_kernel(
    const float* __restrict__ xyz, const float* __restrict__ new_xyz,
    float* __restrict__ grouped) {
  const int w    = threadIdx.x >> 5;
  const int lane = threadIdx.x & 31;
  const int gid  = blockIdx.x * 8 + w;      // centroid id in [0, B*M)
  const int bidx = gid / MM;
  const int midx = gid % MM;

  __shared__ int s_slots[8][NS];

  const float* ctr = new_xyz + ((size_t)bidx * MM + midx) * 3;
  const float cx = ctr[0], cy = ctr[1], cz = ctr[2];
  const float* base = xyz + (size_t)bidx * NN * 3;

  int cnt = 0;
  const unsigned ltm = (1u << lane) - 1u;
  for (int j0 = 0; j0 < NN && cnt < NS; j0 += 32) {
    int j = j0 + lane;
    float dx = base[j * 3 + 0] - cx;
    float dy = base[j * 3 + 1] - cy;
    float dz = base[j * 3 + 2] - cz;
    bool in = (dx * dx + dy * dy + dz * dz) <= R2;
    unsigned msk = (unsigned)__ballot(in);
    int pos = cnt + __popc(msk & ltm);
    if (in && pos < NS) s_slots[w][pos] = j;
    cnt += __popc(msk);
  }

  int sel;
  if (cnt == 0) {
    sel = midx;                    // reference fallback: arange(M) index
  } else {
    sel = s_slots[w][lane % cnt];  // cyclic repeat when cnt < NS
  }
  const float* q = base + (size_t)sel * 3;
  float* o = grouped + (((size_t)bidx * MM + midx) * NS + lane) * 3;
  o[0] = q[0] - cx;
  o[1] = q[1] - cy;
  o[2] = q[2] - cz;
}

// ---------------------------------------------------------------------------
// Kernel 3: per-group MLP with fp32 WMMA (16x16x4) + channel max-pool.
// One wave per centroid. Activations ping-pong between two padded LDS
// buffers (stride 65 floats -> conflict-free fragment loads). All tile (nt)
// loops are ROLLED (#pragma unroll 1) so only one accumulator pair is
// architecturally live at a time -> VGPR use stays < 256 (no s_set_vgpr_msb
// churn); kt gets unroll 4 for ILP between chained WMMAs.
// BN folded to y = gemm*s + t.
// ---------------------------------------------------------------------------
__global__ __launch_bounds__(64)
__attribute__((amdgpu_waves_per_eu(4)))
void mlp_kernel(
    const float* __restrict__ w1, const float* __restrict__ bb1,
    const float* __restrict__ gg1, const float* __restrict__ be1,
    const float* __restrict__ mu1, const float* __restrict__ vv1,
    const float* __restrict__ w2, const float* __restrict__ bb2,
    const float* __restrict__ gg2, const float* __restrict__ be2,
    const float* __restrict__ mu2, const float* __restrict__ vv2,
    const float* __restrict__ w3, const float* __restrict__ bb3,
    const float* __restrict__ gg3, const float* __restrict__ be3,
    const float* __restrict__ mu3, const float* __restrict__ vv3,
    const float* __restrict__ grouped, float* __restrict__ out_pts) {
  // s_aff layout: [0:64) s1 [64:128) t1 [128:192) s2 [192:256) t2
  //               [256:384) s3 [384:512) t3
  __shared__ float s_aff[512];
  __shared__ float s_w1p[64 * 4];        // W1 padded K=3 -> 4 (col 3 = 0)
  __shared__ float s_act0[2][32 * 4];    // padded (x,y,z,0) per sample
  __shared__ float s_actA[2][32 * 65];   // layer-1 output, stride 65
  __shared__ float s_actB[2][32 * 65];   // layer-2 output, stride 65

  for (int c = threadIdx.x; c < 256; c += blockDim.x) {
    int ch, base, tstep;
    const float *G, *BE, *MU, *VA, *BI;
    if (c < 64)       { ch = c;       base = 0;   tstep = 64;  G = gg1; BE = be1; MU = mu1; VA = vv1; BI = bb1; }
    else if (c < 128) { ch = c - 64;  base = 128; tstep = 64;  G = gg2; BE = be2; MU = mu2; VA = vv2; BI = bb2; }
    else              { ch = c - 128; base = 256; tstep = 128; G = gg3; BE = be3; MU = mu3; VA = vv3; BI = bb3; }
    float sc = G[ch] * rsqrtf(VA[ch] + EPSV);
    s_aff[base + ch] = sc;
    s_aff[base + tstep + ch] = (BI[ch] - MU[ch]) * sc + BE[ch];
  }
  for (int i = threadIdx.x; i < 64; i += blockDim.x) {
    s_w1p[i * 4 + 0] = w1[i * 3 + 0];
    s_w1p[i * 4 + 1] = w1[i * 3 + 1];
    s_w1p[i * 4 + 2] = w1[i * 3 + 2];
    s_w1p[i * 4 + 3] = 0.0f;
  }
  __syncthreads();

  const int wave = threadIdx.x >> 5;
  const int lane = threadIdx.x & 31;
  const int lo = lane & 15, hi = lane >> 4;
  float* act0 = &s_act0[wave][0];
  float* actA = &s_actA[wave][0];
  float* actB = &s_actB[wave][0];
  const int nwaves = gridDim.x * 2;

  for (int cent = blockIdx.x * 2 + wave; cent < BB * MM; cent += nwaves) {
    const int bidx = cent / MM;
    const int midx = cent % MM;

    // stage input (lane == sample), pad K to 4 with zero
    {
      const float* gp = grouped + ((size_t)cent * NS + lane) * 3;
      act0[lane * 4 + 0] = gp[0];
      act0[lane * 4 + 1] = gp[1];
      act0[lane * 4 + 2] = gp[2];
      act0[lane * 4 + 3] = 0.0f;
    }

    const int kb0 = hi * 2;  // K offset within a K=4 step for this lane half

    // ---- Layer 1: (32x4) x (4x64) -> actA[32][64] -------------------------
    {
      v2f a0 = {act0[lo * 4 + kb0],        act0[lo * 4 + kb0 + 1]};
      v2f a1 = {act0[(16 + lo) * 4 + kb0], act0[(16 + lo) * 4 + kb0 + 1]};
#pragma unroll 1
      for (int nt = 0; nt < 4; ++nt) {
        int n = nt * 16 + lo;
        v2f bf = {s_w1p[n * 4 + kb0], s_w1p[n * 4 + kb0 + 1]};
        v8f c0 = {}, c1 = {};
        c0 = wmma_k4(a0, bf, c0);
        c1 = wmma_k4(a1, bf, c1);
        float sc = s_aff[n], tb = s_aff[64 + n];
#pragma unroll
        for (int r = 0; r < 8; ++r) {
          float y0 = fmaxf(c0[r] * sc + tb, 0.0f);
          float y1 = fmaxf(c1[r] * sc + tb, 0.0f);
          actA[(hi * 8 + r) * 65 + n]      = y0;
          actA[(16 + hi * 8 + r) * 65 + n] = y1;
        }
      }
    }

    // ---- Layer 2: (32x64) x (64x64) -> actB[32][64] -----------------------
#pragma unroll 1
    for (int nt = 0; nt < 4; ++nt) {
      int n = nt * 16 + lo;
      v8f c0 = {}, c1 = {};
#pragma unroll 4
      for (int kt = 0; kt < 16; ++kt) {
        int kb = kt * 4 + kb0;
        v2f a0 = {actA[lo * 65 + kb],        actA[lo * 65 + kb + 1]};
        v2f a1 = {actA[(16 + lo) * 65 + kb], actA[(16 + lo) * 65 + kb + 1]};
        v2f bf = {w2[n * 64 + kb], w2[n * 64 + kb + 1]};
        c0 = wmma_k4(a0, bf, c0);
        c1 = wmma_k4(a1, bf, c1);
      }
      float sc = s_aff[128 + n], tb = s_aff[192 + n];
#pragma unroll
      for (int r = 0; r < 8; ++r) {
        float y0 = fmaxf(c0[r] * sc + tb, 0.0f);
        float y1 = fmaxf(c1[r] * sc + tb, 0.0f);
        actB[(hi * 8 + r) * 65 + n]      = y0;
        actB[(16 + hi * 8 + r) * 65 + n] = y1;
      }
    }

    // ---- Layer 3: (32x64) x (64x128) + ReLU + max over 32 samples --------
#pragma unroll 1
    for (int nt = 0; nt < 8; ++nt) {
      int n = nt * 16 + lo;
      v8f c0 = {}, c1 = {};
#pragma unroll 4
      for (int kt = 0; kt < 16; ++kt) {
        int kb = kt * 4 + kb0;
        v2f a0 = {actB[lo * 65 + kb],        actB[lo * 65 + kb + 1]};
        v2f a1 = {actB[(16 + lo) * 65 + kb], actB[(16 + lo) * 65 + kb + 1]};
        v2f bf = {w3[n * 64 + kb], w3[n * 64 + kb + 1]};
        c0 = wmma_k4(a0, bf, c0);
        c1 = wmma_k4(a1, bf, c1);
      }
      float sc = s_aff[256 + n], tb = s_aff[384 + n];
      float mx = 0.0f;  // ReLU outputs are >= 0, so 0 is a safe identity
#pragma unroll
      for (int r = 0; r < 8; ++r) {
        mx = fmaxf(mx, fmaxf(c0[r] * sc + tb, 0.0f));
        mx = fmaxf(mx, fmaxf(c1[r] * sc + tb, 0.0f));
      }
      mx = fmaxf(mx, __shfl_xor(mx, 16));  // combine lane halves (M=r vs M=8+r)
      if (lane < 16) {
        int ch = nt * 16 + lane;
        // output layout: (B, 128, M)
        out_pts[((size_t)bidx * 128 + ch) * MM + midx] = mx;
      }
    }
  }
}

// ---------------------------------------------------------------------------
extern "C" void kernel_launch(void* const* d_in, const int* in_sizes, int n_in,
                              void* d_out, int out_size, void* d_ws, size_t ws_size,
                              hipStream_t stream) {
  (void)in_sizes; (void)n_in; (void)out_size; (void)ws_size;
  const float* xyz = (const float*)d_in[0];
  const float* w1  = (const float*)d_in[1];
  const float* b1  = (const float*)d_in[2];
  const float* g1  = (const float*)d_in[3];
  const float* be1 = (const float*)d_in[4];
  const float* m1  = (const float*)d_in[5];
  const float* v1  = (const float*)d_in[6];
  const float* w2  = (const float*)d_in[7];
  const float* b2  = (const float*)d_in[8];
  const float* g2  = (const float*)d_in[9];
  const float* be2 = (const float*)d_in[10];
  const float* m2  = (const float*)d_in[11];
  const float* v2  = (const float*)d_in[12];
  const float* w3  = (const float*)d_in[13];
  const float* b3  = (const float*)d_in[14];
  const float* g3  = (const float*)d_in[15];
  const float* be3 = (const float*)d_in[16];
  const float* m3  = (const float*)d_in[17];
  const float* v3  = (const float*)d_in[18];

  float* out      = (float*)d_out;
  float* new_xyz  = out;                              // (B, M, 3)
  float* out_pts  = out + (size_t)BB * MM * 3;        // (B, 128, M)
  float* grouped  = (float*)d_ws;                     // (B, M, NS, 3)

  fps_kernel<<<BB, 1024, 0, stream>>>(xyz, new_xyz);
  ballgroup_kernel<<<(BB * MM) / 8, 256, 0, stream>>>(xyz, new_xyz, grouped);
  mlp_kernel<<<4096, 64, 0, stream>>>(w1, b1, g1, be1, m1, v1,
                                      w2, b2, g2, be2, m2, v2,
                                      w3, b3, g3, be3, m3, v3,
                                      grouped, out_pts);
}